// VectorQuantizer_24550033063937
// MI455X (gfx1250) — compile-verified
//
#include <hip/hip_runtime.h>
#include <hip/hip_bf16.h>

typedef float v2f __attribute__((ext_vector_type(2)));
typedef float v8f __attribute__((ext_vector_type(8)));

#define N_TOT 65536      // 64 * 32 * 32 vectors
#define K_CB  1024       // codebook size
#define VDIM  64         // code dimension
#define HW    1024       // 32*32
#define CHW   65536      // 64*1024

// ---------------------------------------------------------------------------
// Kernel 1: codebook row norms + zero counts
// ---------------------------------------------------------------------------
__global__ __launch_bounds__(256) void vq_init_kernel(const float* __restrict__ E,
                                                      float* __restrict__ Enorm,
                                                      unsigned int* __restrict__ counts) {
    int k = blockIdx.x * blockDim.x + threadIdx.x;   // 0..1023
    if (k < K_CB) {
        const float* er = E + k * VDIM;
        float s = 0.0f;
#pragma unroll 8
        for (int d = 0; d < VDIM; ++d) { float v = er[d]; s += v * v; }
        Enorm[k]  = s;
        counts[k] = 0u;
    }
}

// ---------------------------------------------------------------------------
// Kernel 2: fused distance GEMM (exact f32 WMMA) + row argmin.
// One wave handles one 16-row tile of Ze; loops over all 1024 codes in
// 16-column chunks; D=64 contraction = 16 chained V_WMMA_F32_16X16X4_F32.
// ---------------------------------------------------------------------------
__global__ __launch_bounds__(256) void vq_argmin_kernel(const float* __restrict__ X,
                                                        const float* __restrict__ E,
                                                        const float* __restrict__ Enorm,
                                                        int* __restrict__ idxOut) {
    const int lane = threadIdx.x & 31;
    const int wave = threadIdx.x >> 5;
    const int tile = blockIdx.x * 8 + wave;          // 16-row tile id (0..4095)
    const int hi   = lane >> 4;                      // 0: lanes 0-15, 1: lanes 16-31
    const int l16  = lane & 15;
    const int row  = tile * 16 + l16;                // this lane's A-matrix row (M)
    const int off  = hi * 2;                         // d%4 offset per ISA A layout

    // Ze[row][d] lives at X[(row>>10)*CHW + (row&1023) + d*HW]  ([B,C,H,W] layout)
    const int rbase = (row >> 10) * CHW + (row & (HW - 1));

    // Resident A-tile: 32 floats/lane. Lane<16 holds d%4 in {0,1}, lane>=16 {2,3}.
    float A[32];
#pragma unroll
    for (int kk = 0; kk < 16; ++kk) {
        A[2 * kk + 0] = X[rbase + (4 * kk + off + 0) * HW];
        A[2 * kk + 1] = X[rbase + (4 * kk + off + 1) * HW];
    }

    float best[8];
    int   bidx[8];
#pragma unroll
    for (int r = 0; r < 8; ++r) { best[r] = 3.0e38f; bidx[r] = 0x7fffffff; }

    for (int cb = 0; cb < K_CB; cb += 16) {
        const int   code = cb + l16;                 // this lane's B-matrix code row
        const float en   = Enorm[code];              // == Enorm of this lane's C column
        const float* erow = E + code * VDIM + off;   // 8-byte aligned (off even)

        v8f acc = {};
#pragma unroll
        for (int kk = 0; kk < 16; ++kk) {
            v2f a; a.x = A[2 * kk]; a.y = A[2 * kk + 1];
            v2f b = *(const v2f*)(erow + 4 * kk);    // E[code][4kk+off], +off+1
            acc = __builtin_amdgcn_wmma_f32_16x16x4_f32(
                false, a, false, b, (short)0, acc, false, false);
        }
        // element (VGPR r, this lane): row = tile*16 + 8*hi + r, col = cb + l16
        // dist = ||E||^2 - 2*dot  (||Ze||^2 constant per row; irrelevant to argmin)
#pragma unroll
        for (int r = 0; r < 8; ++r) {
            float d  = en - 2.0f * acc[r];
            int   ci = cb + l16;
            if (d < best[r] || (d == best[r] && ci < bidx[r])) { best[r] = d; bidx[r] = ci; }
        }
    }

    // Min-reduce across the 16 lanes of each half (cols); keep argmin, first-index ties.
#pragma unroll
    for (int m = 8; m >= 1; m >>= 1) {
#pragma unroll
        for (int r = 0; r < 8; ++r) {
            float ob = __shfl_xor(best[r], m, 32);
            int   oi = __shfl_xor(bidx[r], m, 32);
            if (ob < best[r] || (ob == best[r] && oi < bidx[r])) { best[r] = ob; bidx[r] = oi; }
        }
    }
    if (l16 == 0) {
#pragma unroll
        for (int r = 0; r < 8; ++r)
            idxOut[tile * 16 + hi * 8 + r] = bidx[r];
    }
}

// ---------------------------------------------------------------------------
// Kernel 3: gather Zq = E[idx] back to [B,C,H,W], squared-error partials,
// usage histogram. One thread per vector n; hw-coalesced stores.
// ---------------------------------------------------------------------------
__global__ __launch_bounds__(256) void vq_gather_kernel(const float* __restrict__ X,
                                                        const float* __restrict__ E,
                                                        const int* __restrict__ idx,
                                                        float* __restrict__ Zq,
                                                        float* __restrict__ partial,
                                                        unsigned int* __restrict__ counts) {
    const int n = blockIdx.x * blockDim.x + threadIdx.x;   // 0..65535
    const int k = idx[n];
    atomicAdd(&counts[k], 1u);

    const float* er  = E + k * VDIM;
    const int   base = (n >> 10) * CHW + (n & (HW - 1));
    float s = 0.0f;
#pragma unroll 8
    for (int d = 0; d < VDIM; ++d) {
        float zq = er[d];
        float ze = X[base + d * HW];
        float df = zq - ze;
        s += df * df;
        Zq[base + d * HW] = zq;
    }

    __shared__ float sm[256];
    sm[threadIdx.x] = s;
    __syncthreads();
    for (int st = 128; st > 0; st >>= 1) {
        if (threadIdx.x < st) sm[threadIdx.x] += sm[threadIdx.x + st];
        __syncthreads();
    }
    if (threadIdx.x == 0) partial[blockIdx.x] = sm[0];
}

// ---------------------------------------------------------------------------
// Kernel 4: entropy -> perplexity, final scalars.
// ---------------------------------------------------------------------------
__global__ __launch_bounds__(1024) void vq_finalize_kernel(const unsigned int* __restrict__ counts,
                                                           const float* __restrict__ partial,
                                                           float* __restrict__ out) {
    __shared__ float sm[1024];
    const int t = threadIdx.x;
    float p = (float)counts[t] * (1.0f / (float)N_TOT);
    sm[t] = -p * log2f(p + 1e-10f);
    __syncthreads();
    for (int st = 512; st > 0; st >>= 1) {
        if (t < st) sm[t] += sm[t + st];
        __syncthreads();
    }
    if (t == 0) {
        float entropy = sm[0];
        float ss = 0.0f;
        for (int i = 0; i < 256; ++i) ss += partial[i];   // fixed order: deterministic
        float mse = ss * (1.0f / (float)(N_TOT * VDIM));
        out[0]       = mse + 0.25f * mse;   // q_latent + BETA * e_latent
        out[4194305] = mse;                 // e_latent_loss
        out[4194306] = mse;                 // q_latent_loss
        out[4194307] = exp2f(entropy);      // perplexity
    }
}

// ---------------------------------------------------------------------------
extern "C" void kernel_launch(void* const* d_in, const int* in_sizes, int n_in,
                              void* d_out, int out_size, void* d_ws, size_t ws_size,
                              hipStream_t stream) {
    const float* X = (const float*)d_in[0];   // [64,64,32,32] f32
    const float* E = (const float*)d_in[1];   // [1024,64] f32
    float* out = (float*)d_out;               // [loss, Zq(4194304), e, q, perp]

    // workspace layout
    float*        Enorm   = (float*)d_ws;                              // 1024 f32
    unsigned int* counts  = (unsigned int*)((char*)d_ws + 1024 * 4);   // 1024 u32
    float*        partial = (float*)((char*)d_ws + 2048 * 4);          // 256 f32
    int*          idx     = (int*)((char*)d_ws + (2048 + 256) * 4);    // 65536 i32

    vq_init_kernel  <<<4,    256, 0, stream>>>(E, Enorm, counts);
    vq_argmin_kernel<<<512,  256, 0, stream>>>(X, E, Enorm, idx);      // 4096 tiles
    vq_gather_kernel<<<256,  256, 0, stream>>>(X, E, idx, out + 1, partial, counts);
    vq_finalize_kernel<<<1, 1024, 0, stream>>>(counts, partial, out);
}